// GIN_52699248722373
// MI455X (gfx1250) — compile-verified
//
#include <hip/hip_runtime.h>
#include <hip/hip_bf16.h>
#include <math.h>

typedef __attribute__((ext_vector_type(2))) float v2f;
typedef __attribute__((ext_vector_type(8))) float v8f;

#define N_NODES 100000
#define N_EDGES 1600000
#define IN_DIM  64
#define HID_DIM 128
#define OUT_DIM 40

// ---------------------------------------------------------------------------
// Edge-parallel scatter-add: agg[dst[e]] += feat[src[e]]  (D = 64 or 128)
// One thread handles one float4 chunk of one edge. Gathers hit L2 (feat fits
// in the 192MB L2); atomics are hardware global_atomic_add_f32 in L2.
// ---------------------------------------------------------------------------
template <int D>
__global__ __launch_bounds__(256) void scatter_add_kernel(
    const float* __restrict__ feat, const int* __restrict__ src,
    const int* __restrict__ dst, float* __restrict__ agg, int n_edges) {
  constexpr int CH = D / 4;  // float4 chunks per edge (16 or 32)
  long long tid = (long long)blockIdx.x * 256 + threadIdx.x;
  int e = (int)(tid / CH);
  int c = (int)(tid % CH);
  if (e >= n_edges) return;
  int s = src[e];
  int d = dst[e];
  const float4 v = *(const float4*)(feat + (long long)s * D + c * 4);
  float* p = agg + (long long)d * D + c * 4;
  unsafeAtomicAdd(p + 0, v.x);
  unsafeAtomicAdd(p + 1, v.y);
  unsafeAtomicAdd(p + 2, v.z);
  unsafeAtomicAdd(p + 3, v.w);
}

// ---------------------------------------------------------------------------
// Fused MLP1: h = relu(relu(agg @ W1 + b1) @ W2 + b2)
// Block = 256 threads (8 waves), 16 nodes per block (100000 = 6250*16 exact).
// Wave w owns output columns [16w, 16w+16). fp32 WMMA 16x16x4.
// ---------------------------------------------------------------------------
__global__ __launch_bounds__(256) void gin_mlp1_kernel(
    const float* __restrict__ agg,  // [N,64]
    const float* __restrict__ W1,   // [64,128]
    const float* __restrict__ b1,   // [128]
    const float* __restrict__ W2,   // [128,128]
    const float* __restrict__ b2,   // [128]
    float* __restrict__ h)          // [N,128]
{
  __shared__ float A_s[16 * 65];    // 16x64 tile, padded (bank-conflict free)
  __shared__ float H_s[16 * 129];   // 16x128 intermediate, padded

  const int tid  = threadIdx.x;
  const int wave = tid >> 5;
  const int lane = tid & 31;
  const int node0 = blockIdx.x * 16;

  // cooperative load of the 16x64 A tile (coalesced rows)
  for (int i = tid; i < 16 * IN_DIM; i += 256) {
    int r = i >> 6, c = i & 63;
    A_s[r * 65 + c] = agg[(long long)(node0 + r) * IN_DIM + c];
  }
  __syncthreads();

  const int nloc  = lane & 15;            // N (or M for A-frag) within tile
  const int khalf = (lane >> 4) * 2;      // lanes 16-31 hold K+2 per ISA layout
  const int mbase = (lane >> 4) * 8;      // C/D: upper half-wave holds M+8
  const int ncol  = wave * 16 + nloc;     // this wave's output column

  // ---- GEMM1: (16x64) x (64x128-sliced-16) ----
  v8f acc = {0.f, 0.f, 0.f, 0.f, 0.f, 0.f, 0.f, 0.f};
  for (int k0 = 0; k0 < IN_DIM; k0 += 4) {
    const int k = k0 + khalf;
    v2f a;
    a.x = A_s[nloc * 65 + k];
    a.y = A_s[nloc * 65 + k + 1];
    v2f b;
    b.x = W1[k * HID_DIM + ncol];
    b.y = W1[(k + 1) * HID_DIM + ncol];
    acc = __builtin_amdgcn_wmma_f32_16x16x4_f32(false, a, false, b,
                                                (short)0, acc, false, false);
  }
  {
    const float bias = b1[ncol];
#pragma unroll
    for (int r = 0; r < 8; ++r) {
      float v = acc[r] + bias;
      H_s[(mbase + r) * 129 + ncol] = v > 0.f ? v : 0.f;
    }
  }
  __syncthreads();

  // ---- GEMM2: (16x128) x (128x128-sliced-16) ----
  v8f acc2 = {0.f, 0.f, 0.f, 0.f, 0.f, 0.f, 0.f, 0.f};
  for (int k0 = 0; k0 < HID_DIM; k0 += 4) {
    const int k = k0 + khalf;
    v2f a;
    a.x = H_s[nloc * 129 + k];
    a.y = H_s[nloc * 129 + k + 1];
    v2f b;
    b.x = W2[k * HID_DIM + ncol];
    b.y = W2[(k + 1) * HID_DIM + ncol];
    acc2 = __builtin_amdgcn_wmma_f32_16x16x4_f32(false, a, false, b,
                                                 (short)0, acc2, false, false);
  }
  {
    const float bias = b2[ncol];
#pragma unroll
    for (int r = 0; r < 8; ++r) {
      float v = acc2[r] + bias;
      v = v > 0.f ? v : 0.f;
      h[(long long)(node0 + mbase + r) * HID_DIM + ncol] = v;
    }
  }
}

// ---------------------------------------------------------------------------
// MLP2: logits = relu(agg2 @ W3 + b3), 40 output cols as 3 masked 16-wide
// WMMA tiles. Block = 128 threads (4 waves), 64 nodes per block; wave w owns
// nodes [16w,16w+16) of the block.
// ---------------------------------------------------------------------------
__global__ __launch_bounds__(128) void gin_mlp2_kernel(
    const float* __restrict__ agg2,  // [N,128]
    const float* __restrict__ W3,    // [128,40]
    const float* __restrict__ b3,    // [40]
    float* __restrict__ out,         // [N,40]
    int n_nodes) {
  __shared__ float A_s[64 * 129];  // 64x128 tile, padded

  const int tid  = threadIdx.x;
  const int wave = tid >> 5;
  const int lane = tid & 31;
  const int node0 = blockIdx.x * 64;

  for (int i = tid; i < 64 * HID_DIM; i += 128) {
    int r = i >> 7, c = i & 127;
    int node = node0 + r;
    A_s[r * 129 + c] =
        (node < n_nodes) ? agg2[(long long)node * HID_DIM + c] : 0.f;
  }
  __syncthreads();

  const int nloc  = lane & 15;
  const int khalf = (lane >> 4) * 2;
  const int mbase = (lane >> 4) * 8;
  const int arow  = wave * 16 + nloc;  // A row within the block tile

  v8f acc[3];
#pragma unroll
  for (int t = 0; t < 3; ++t) acc[t] = (v8f){0.f, 0.f, 0.f, 0.f, 0.f, 0.f, 0.f, 0.f};

  for (int k0 = 0; k0 < HID_DIM; k0 += 4) {
    const int k = k0 + khalf;
    v2f a;
    a.x = A_s[arow * 129 + k];
    a.y = A_s[arow * 129 + k + 1];
#pragma unroll
    for (int t = 0; t < 3; ++t) {
      const int col = t * 16 + nloc;
      v2f b;
      b.x = (col < OUT_DIM) ? W3[k * OUT_DIM + col] : 0.f;
      b.y = (col < OUT_DIM) ? W3[(k + 1) * OUT_DIM + col] : 0.f;
      acc[t] = __builtin_amdgcn_wmma_f32_16x16x4_f32(false, a, false, b,
                                                     (short)0, acc[t], false,
                                                     false);
    }
  }

#pragma unroll
  for (int t = 0; t < 3; ++t) {
    const int col = t * 16 + nloc;
    if (col < OUT_DIM) {
      const float bias = b3[col];
#pragma unroll
      for (int r = 0; r < 8; ++r) {
        int node = node0 + wave * 16 + mbase + r;
        if (node < n_nodes) {
          float v = acc[t][r] + bias;
          v = v > 0.f ? v : 0.f;
          out[(long long)node * OUT_DIM + col] = v;
        }
      }
    }
  }
}

// ---------------------------------------------------------------------------
// In-place row-wise log_softmax over 40 classes; one thread per node.
// ---------------------------------------------------------------------------
__global__ __launch_bounds__(256) void log_softmax40_kernel(float* __restrict__ out,
                                                            int n_nodes) {
  int node = blockIdx.x * 256 + threadIdx.x;
  if (node >= n_nodes) return;
  float* row = out + (long long)node * OUT_DIM;
  float v[OUT_DIM];
  float m = -INFINITY;
#pragma unroll
  for (int i = 0; i < OUT_DIM; ++i) {
    v[i] = row[i];
    m = fmaxf(m, v[i]);
  }
  float s = 0.f;
#pragma unroll
  for (int i = 0; i < OUT_DIM; ++i) s += expf(v[i] - m);
  const float ls = logf(s);
#pragma unroll
  for (int i = 0; i < OUT_DIM; ++i) row[i] = v[i] - m - ls;
}

// ---------------------------------------------------------------------------
extern "C" void kernel_launch(void* const* d_in, const int* in_sizes, int n_in,
                              void* d_out, int out_size, void* d_ws,
                              size_t ws_size, hipStream_t stream) {
  (void)in_sizes; (void)n_in; (void)out_size; (void)ws_size;
  const float* x  = (const float*)d_in[0];
  const int*   ei = (const int*)d_in[1];  // [2, E] edge index (int32 per harness)
  const float* W1 = (const float*)d_in[2];
  const float* b1 = (const float*)d_in[3];
  const float* W2 = (const float*)d_in[4];
  const float* b2 = (const float*)d_in[5];
  const float* W3 = (const float*)d_in[6];
  const float* b3 = (const float*)d_in[7];
  float* out = (float*)d_out;

  const int* src = ei;
  const int* dst = ei + N_EDGES;

  // Workspace layout: agg1 [N,64] | h [N,128] | agg2 [N,128]  (~128 MB total)
  float* agg1 = (float*)d_ws;
  float* h    = agg1 + (size_t)N_NODES * IN_DIM;
  float* agg2 = h + (size_t)N_NODES * HID_DIM;

  // ---- conv1 aggregate: agg1 = x + scatter_add(x[src] -> dst) ----
  hipMemcpyAsync(agg1, x, (size_t)N_NODES * IN_DIM * sizeof(float),
                 hipMemcpyDeviceToDevice, stream);
  {
    long long total = (long long)N_EDGES * (IN_DIM / 4);
    int blocks = (int)((total + 255) / 256);
    scatter_add_kernel<IN_DIM><<<blocks, 256, 0, stream>>>(x, src, dst, agg1,
                                                           N_EDGES);
  }

  // ---- conv1 MLP (fused Linear+ReLU+Linear+ReLU) ----
  gin_mlp1_kernel<<<N_NODES / 16, 256, 0, stream>>>(agg1, W1, b1, W2, b2, h);

  // ---- conv2 aggregate: agg2 = h + scatter_add(h[src] -> dst) ----
  hipMemcpyAsync(agg2, h, (size_t)N_NODES * HID_DIM * sizeof(float),
                 hipMemcpyDeviceToDevice, stream);
  {
    long long total = (long long)N_EDGES * (HID_DIM / 4);
    int blocks = (int)((total + 255) / 256);
    scatter_add_kernel<HID_DIM><<<blocks, 256, 0, stream>>>(h, src, dst, agg2,
                                                            N_EDGES);
  }

  // ---- conv2 MLP (Linear+ReLU) -> d_out, then in-place log_softmax ----
  gin_mlp2_kernel<<<(N_NODES + 63) / 64, 128, 0, stream>>>(agg2, W3, b3, out,
                                                           N_NODES);
  log_softmax40_kernel<<<(N_NODES + 255) / 256, 256, 0, stream>>>(out, N_NODES);
}